// TripletLoss_58488864637169
// MI455X (gfx1250) — compile-verified
//
#include <hip/hip_runtime.h>
#include <math.h>
#include <stdint.h>

typedef float v2f __attribute__((ext_vector_type(2)));
typedef float v8f __attribute__((ext_vector_type(8)));
typedef unsigned int u32x4 __attribute__((ext_vector_type(4)));
typedef int i32x8 __attribute__((ext_vector_type(8)));
typedef int i32x4 __attribute__((ext_vector_type(4)));

#define BT 128   // block tile edge (M and N)
#define KC 32    // K chunk (dwords per LDS row before padding)
#define LDST 34  // LDS row stride in dwords (KC + 2 pad dwords from TDM pad)

// ---------------------------------------------------------------------------
// Kernel 1: per-row squared norms. One wave32 per row.
// ---------------------------------------------------------------------------
__global__ void sqnorm_kernel(const float* __restrict__ A,
                              float* __restrict__ sq, int n, int d) {
    const int wave = (blockIdx.x * blockDim.x + threadIdx.x) >> 5;
    const int lane = threadIdx.x & 31;
    if (wave >= n) return;
    const float* row = A + (size_t)wave * d;
    float s = 0.0f;
    for (int j = lane; j < d; j += 32) {
        float v = row[j];
        s += v * v;
    }
    #pragma unroll
    for (int ofs = 16; ofs > 0; ofs >>= 1)
        s += __shfl_down(s, ofs, 32);
    if (lane == 0) sq[wave] = s;
}

// ---------------------------------------------------------------------------
// Kernel 2: 128x128 dist tile per block. TDM stages A/B panels into LDS
// (with TDM pad -> stride-34 rows, bank-conflict-free), 8 waves each compute
// a 16x128 strip with V_WMMA_F32_16X16X4_F32 (A frag reused across 8 subtiles).
// ---------------------------------------------------------------------------
__global__ __launch_bounds__(256)
void dist_wmma_kernel(const float* __restrict__ A,
                      const float* __restrict__ sq,
                      float* __restrict__ dist, int n, int d) {
    __shared__ float As[BT * LDST];  // 17 KB
    __shared__ float Bs[BT * LDST];  // 17 KB

    const int tid  = threadIdx.x;
    const int lane = tid & 31;
    const int wv   = tid >> 5;            // 0..7
    const int bm   = blockIdx.y * BT;     // row block
    const int bn   = blockIdx.x * BT;     // col block

    const int lr   = lane & 15;           // row-within-16 for frag loads / N col
    const int kOff = (lane >> 4) << 1;    // 0 or 2 (K pair per half-wave)

    // TDM descriptor group 1 (tile shape), constant across chunks:
    //   data_size=4B (2), pad_enable, pad_interval=4 (every 32 dwords),
    //   pad_amount=1 (2 dwords), tensor_dim0=d, tensor_dim1=n,
    //   tile_dim0=KC, tile_dim1=BT, tensor_dim0_stride=d.
    i32x8 g1 = {};
    g1[0] = (int)((2u << 16) | (1u << 20) | (4u << 22) | (1u << 25));
    g1[1] = (int)(((unsigned)d & 0xFFFFu) << 16);
    g1[2] = (int)((((unsigned)d >> 16) & 0xFFFFu) | (((unsigned)n & 0xFFFFu) << 16));
    g1[3] = (int)((((unsigned)n >> 16) & 0xFFFFu) | ((unsigned)KC << 16));
    g1[4] = (int)(unsigned)BT;            // tile_dim1 = 128, tile_dim2 = 0
    g1[5] = (int)(unsigned)d;             // tensor_dim0_stride[31:0]
    g1[6] = 0;                            // stride[47:32]=0, dim1_stride lo16=0
    g1[7] = 0;
    const i32x4 gz4 = {};
    const i32x8 gz8 = {};

    const unsigned ldsA = (unsigned)(uintptr_t)(void*)As;  // LDS byte offset
    const unsigned ldsB = (unsigned)(uintptr_t)(void*)Bs;

    // Per-wave LDS base for its A rows (rows bm+16*wv .. +15)
    const float* aBase = &As[(16 * wv + lr) * LDST + kOff];

    v8f acc0 = {}, acc1 = {}, acc2 = {}, acc3 = {};
    v8f acc4 = {}, acc5 = {}, acc6 = {}, acc7 = {};

    for (int k = 0; k < d; k += KC) {
        if (wv == 0) {
            const uint64_t gaA = (uint64_t)(uintptr_t)(A + (size_t)bm * d + k);
            const uint64_t gaB = (uint64_t)(uintptr_t)(A + (size_t)bn * d + k);
            u32x4 g0a = {};
            g0a[0] = 1u;                              // count=1 valid descriptor
            g0a[1] = ldsA;
            g0a[2] = (unsigned)gaA;
            g0a[3] = (unsigned)(gaA >> 32) | 0x80000000u;  // type=2 ("image")
            u32x4 g0b = {};
            g0b[0] = 1u;
            g0b[1] = ldsB;
            g0b[2] = (unsigned)gaB;
            g0b[3] = (unsigned)(gaB >> 32) | 0x80000000u;
            __builtin_amdgcn_tensor_load_to_lds(g0a, g1, gz4, gz4, gz8, 0);
            __builtin_amdgcn_tensor_load_to_lds(g0b, g1, gz4, gz4, gz8, 0);
            __builtin_amdgcn_s_wait_tensorcnt(0);
        }
        __syncthreads();   // LDS panels ready

        #pragma unroll
        for (int kk = 0; kk < KC; kk += 4) {
            const v2f a = *(const v2f*)(aBase + kk);
            const float* bBase = &Bs[lr * LDST + kOff + kk];
            v2f b;
            b = *(const v2f*)(bBase + 0 * 16 * LDST);
            acc0 = __builtin_amdgcn_wmma_f32_16x16x4_f32(false, a, false, b, (short)0, acc0, false, false);
            b = *(const v2f*)(bBase + 1 * 16 * LDST);
            acc1 = __builtin_amdgcn_wmma_f32_16x16x4_f32(false, a, false, b, (short)0, acc1, false, false);
            b = *(const v2f*)(bBase + 2 * 16 * LDST);
            acc2 = __builtin_amdgcn_wmma_f32_16x16x4_f32(false, a, false, b, (short)0, acc2, false, false);
            b = *(const v2f*)(bBase + 3 * 16 * LDST);
            acc3 = __builtin_amdgcn_wmma_f32_16x16x4_f32(false, a, false, b, (short)0, acc3, false, false);
            b = *(const v2f*)(bBase + 4 * 16 * LDST);
            acc4 = __builtin_amdgcn_wmma_f32_16x16x4_f32(false, a, false, b, (short)0, acc4, false, false);
            b = *(const v2f*)(bBase + 5 * 16 * LDST);
            acc5 = __builtin_amdgcn_wmma_f32_16x16x4_f32(false, a, false, b, (short)0, acc5, false, false);
            b = *(const v2f*)(bBase + 6 * 16 * LDST);
            acc6 = __builtin_amdgcn_wmma_f32_16x16x4_f32(false, a, false, b, (short)0, acc6, false, false);
            b = *(const v2f*)(bBase + 7 * 16 * LDST);
            acc7 = __builtin_amdgcn_wmma_f32_16x16x4_f32(false, a, false, b, (short)0, acc7, false, false);
        }
        __syncthreads();   // all waves done reading before next TDM overwrite
    }

    // Epilogue: dist = sqrt(max(sq_m + sq_n - 2*dot, 1e-12))
    const int   mbase = bm + 16 * wv + ((lane >> 4) << 3);
    const float sqm[8] = { sq[mbase+0], sq[mbase+1], sq[mbase+2], sq[mbase+3],
                           sq[mbase+4], sq[mbase+5], sq[mbase+6], sq[mbase+7] };
    v8f* accs[8] = { &acc0, &acc1, &acc2, &acc3, &acc4, &acc5, &acc6, &acc7 };
    #pragma unroll
    for (int t = 0; t < 8; ++t) {
        const int   nn  = bn + 16 * t + lr;
        const float sqn = sq[nn];
        const v8f   a   = *accs[t];
        #pragma unroll
        for (int r = 0; r < 8; ++r) {
            const int mm = mbase + r;
            float dd = sqm[r] + sqn - 2.0f * a[r];
            dd = sqrtf(fmaxf(dd, 1e-12f));
            dist[(size_t)mm * n + nn] = dd;
        }
    }
}

// ---------------------------------------------------------------------------
// Kernel 3: per-row masked max/min + first-index-of-exact-min.
// One 256-thread block per row; row staged in LDS.
// ---------------------------------------------------------------------------
__global__ void rowreduce_kernel(const float* __restrict__ dist,
                                 const int* __restrict__ targets,
                                 const int* __restrict__ flags,
                                 float* __restrict__ same_ap,
                                 float* __restrict__ same_an,
                                 float* __restrict__ dif_ap,
                                 float* __restrict__ dif_an,
                                 int* __restrict__ same_idx,
                                 int* __restrict__ dif_idx, int n) {
    const int i   = blockIdx.x;
    const int tid = threadIdx.x;
    const int nt  = blockDim.x;

    __shared__ float srow[4096];
    __shared__ float r0[256], r1[256], r2[256], r3[256];
    __shared__ int   iA[256], iB[256];
    __shared__ float bc[2];

    const float* drow = dist + (size_t)i * n;
    const int ti = targets[i];
    const int fi = flags[i];

    float mx_shp = -INFINITY, mn_shn = INFINITY;
    float mx_dhp = -INFINITY, mn_dhn = INFINITY;
    for (int j = tid; j < n; j += nt) {
        const float dv = drow[j];
        srow[j] = dv;
        const bool sid = (targets[j] == ti);
        const bool sfl = (flags[j] == fi);
        if (sid && sfl)   mx_shp = fmaxf(mx_shp, dv);
        if (!sid && sfl)  mn_shn = fminf(mn_shn, dv);
        if (sid && !sfl)  mx_dhp = fmaxf(mx_dhp, dv);
        if (!sid && !sfl) mn_dhn = fminf(mn_dhn, dv);
    }
    r0[tid] = mx_shp; r1[tid] = mn_shn; r2[tid] = mx_dhp; r3[tid] = mn_dhn;
    __syncthreads();
    for (int s = nt >> 1; s > 0; s >>= 1) {
        if (tid < s) {
            r0[tid] = fmaxf(r0[tid], r0[tid + s]);
            r1[tid] = fminf(r1[tid], r1[tid + s]);
            r2[tid] = fmaxf(r2[tid], r2[tid + s]);
            r3[tid] = fminf(r3[tid], r3[tid + s]);
        }
        __syncthreads();
    }
    if (tid == 0) {
        same_ap[i] = r0[0];
        same_an[i] = r1[0];
        dif_ap[i]  = r2[0];
        dif_an[i]  = r3[0];
        bc[0] = r1[0];
        bc[1] = r3[0];
    }
    __syncthreads();

    const float mnS = bc[0];
    const float mnD = bc[1];
    int idxS = 0x7fffffff, idxD = 0x7fffffff;
    for (int j = tid; j < n; j += nt) {
        const float dv = srow[j];
        if (dv == mnS && j < idxS) idxS = j;
        if (dv == mnD && j < idxD) idxD = j;
    }
    iA[tid] = idxS; iB[tid] = idxD;
    __syncthreads();
    for (int s = nt >> 1; s > 0; s >>= 1) {
        if (tid < s) {
            iA[tid] = min(iA[tid], iA[tid + s]);
            iB[tid] = min(iB[tid], iB[tid + s]);
        }
        __syncthreads();
    }
    if (tid == 0) {
        same_idx[i] = (iA[0] == 0x7fffffff) ? 0 : iA[0];
        dif_idx[i]  = (iB[0] == 0x7fffffff) ? 0 : iB[0];
    }
}

// ---------------------------------------------------------------------------
// Kernel 4: final losses. Single block, deterministic tree reduction.
// ---------------------------------------------------------------------------
__global__ void loss_kernel(const float* __restrict__ dist,
                            const float* __restrict__ same_ap,
                            const float* __restrict__ same_an,
                            const float* __restrict__ dif_ap,
                            const float* __restrict__ dif_an,
                            const int* __restrict__ same_idx,
                            const int* __restrict__ dif_idx,
                            float* __restrict__ out, int n) {
    const int tid = threadIdx.x;
    __shared__ float red[1024];
    float s = 0.0f;
    for (int i = tid; i < n; i += blockDim.x) {
        const float ap   = same_ap[i];
        const float t1   = ap - same_an[i] + 0.8f;
        const float t2   = dif_ap[i] - dif_an[i] + 0.5f;
        const float dneg = dist[(size_t)same_idx[i] * n + dif_idx[i]];
        const float t3   = ap - dneg + 0.3f;
        s += fmaxf(t1, 0.0f) + fmaxf(t2, 0.0f) + fmaxf(t3, 0.0f);
    }
    red[tid] = s;
    __syncthreads();
    for (int st = blockDim.x >> 1; st > 0; st >>= 1) {
        if (tid < st) red[tid] += red[tid + st];
        __syncthreads();
    }
    if (tid == 0) out[0] = red[0] / (float)n;
}

// ---------------------------------------------------------------------------
extern "C" void kernel_launch(void* const* d_in, const int* in_sizes, int n_in,
                              void* d_out, int out_size, void* d_ws, size_t ws_size,
                              hipStream_t stream) {
    const float* A       = (const float*)d_in[0];
    const int*   targets = (const int*)d_in[1];
    const int*   flags   = (const int*)d_in[2];
    const int n = in_sizes[1];           // 4096
    const int d = in_sizes[0] / n;       // 1024

    float* dist     = (float*)d_ws;                  // n*n
    float* sq       = dist + (size_t)n * n;          // n
    float* same_ap  = sq + n;
    float* same_an  = same_ap + n;
    float* dif_ap   = same_an + n;
    float* dif_an   = dif_ap + n;
    int*   same_idx = (int*)(dif_an + n);
    int*   dif_idx  = same_idx + n;

    sqnorm_kernel<<<(n + 7) / 8, 256, 0, stream>>>(A, sq, n, d);

    dim3 grid(n / BT, n / BT);
    dist_wmma_kernel<<<grid, 256, 0, stream>>>(A, sq, dist, n, d);

    rowreduce_kernel<<<n, 256, 0, stream>>>(dist, targets, flags,
                                            same_ap, same_an, dif_ap, dif_an,
                                            same_idx, dif_idx, n);

    loss_kernel<<<1, 1024, 0, stream>>>(dist, same_ap, same_an, dif_ap, dif_an,
                                        same_idx, dif_idx, (float*)d_out, n);
}